// KWinnersTakeAll_52905407152509
// MI455X (gfx1250) — compile-verified
//
#include <hip/hip_runtime.h>
#include <stdint.h>

// k-winners-take-all, B=4096 rows x E=8192 cols fp32, k=ceil(0.05*8192)=410.
// One workgroup per row. Row staged global->LDS and mask streamed LDS->global
// via the gfx1250 Tensor Data Mover (TENSOR_LOAD_TO_LDS / TENSOR_STORE_FROM_LDS,
// TENSORcnt). 4-pass 8-bit radix select in LDS finds the exact k-th largest
// key; exact lower-index-first tie handling matches lax.top_k. Memory floor
// ~11us (256MB @ 23.3TB/s); all selection work is LDS-local.

#define KWTA_E   8192
#define KWTA_K   410
#define KWTA_T   256
#define KWTA_PT  (KWTA_E / KWTA_T)   // 32 elements per thread

typedef unsigned int kw_u32x4 __attribute__((ext_vector_type(4)));
typedef int          kw_i32x8 __attribute__((ext_vector_type(8)));
typedef int          kw_i32x4 __attribute__((ext_vector_type(4)));

#if defined(__HIP_DEVICE_COMPILE__) && __has_builtin(__builtin_amdgcn_tensor_load_to_lds)
#define KWTA_TDM_LOAD 1
#endif
#if defined(__HIP_DEVICE_COMPILE__) && __has_builtin(__builtin_amdgcn_tensor_store_from_lds)
#define KWTA_TDM_STORE 1
#endif
// amdgpu-toolchain (therock-10.0 headers) ships this header and uses the
// 6-arg builtin; ROCm 7.2 has the 5-arg form (per probe).
#if __has_include(<hip/amd_detail/amd_gfx1250_TDM.h>)
#define KWTA_TDM_6ARG 1
#endif

__device__ __forceinline__ void kwta_wait_tensorcnt0() {
#if defined(__HIP_DEVICE_COMPILE__)
#if __has_builtin(__builtin_amdgcn_s_wait_tensorcnt)
  __builtin_amdgcn_s_wait_tensorcnt(0);
#else
  asm volatile("s_wait_tensorcnt 0" ::: "memory");
#endif
#endif
}

// Build D# group0: count=1 descriptor, LDS byte offset, 57-bit global addr, type=2.
__device__ __forceinline__ kw_u32x4 kwta_desc_g0(uint32_t lds_off, uint64_t gaddr) {
  kw_u32x4 g0;
  g0.x = 1u;                                                 // count=1, is_restore=0
  g0.y = lds_off;                                            // lds_addr [63:32]
  g0.z = (uint32_t)gaddr;                                    // global_addr lo
  g0.w = ((uint32_t)(gaddr >> 32) & 0x01FFFFFFu) | (2u << 30); // addr[56:32] | type=2
  return g0;
}

// D# group1 for a 1-row tile of KWTA_E fp32 elements.
__device__ __forceinline__ kw_i32x8 kwta_desc_g1() {
  kw_i32x8 g1;
  g1[0] = 0x00020000;          // workgroup_mask=0, data_size=2 (4B), no pad/iterate
  g1[1] = (int)(0x2000u << 16); // tensor_dim0[15:0]=8192 in bits[63:48]
  g1[2] = 0x00010000;          // tensor_dim1=1 (bits[95:80])
  g1[3] = (int)(0x2000u << 16); // tile_dim0=8192 (bits[127:112])
  g1[4] = 0x00000001;          // tile_dim1=1, tile_dim2=0
  g1[5] = 0x00002000;          // tensor_dim0_stride=8192 (bits[191:160])
  g1[6] = 0;                   // stride hi / tensor_dim1_stride lo
  g1[7] = 0;
  return g1;
}

__global__ __launch_bounds__(KWTA_T) void kwta_kernel(const float* __restrict__ x,
                                                      float* __restrict__ out,
                                                      int nrows) {
  __shared__ uint32_t keys[KWTA_E];     // 32KB: raw row bits -> order keys -> mask bits
  __shared__ uint32_t hist[256];        // radix histogram
  __shared__ uint32_t tcnt[KWTA_T];     // per-thread tie counts (exclusive scanned)
  __shared__ uint32_t sh_prefix;
  __shared__ uint32_t sh_remain;

  const int row = blockIdx.x;
  if (row >= nrows) return;
  const int tid = threadIdx.x;
  const float* rowp = x + (size_t)row * KWTA_E;
  const uint32_t keys_lds_off = (uint32_t)(uintptr_t)(&keys[0]); // AS3 offset in low 32 bits

  // ---- Stage one row (32KB) into LDS via TDM (wave 0 issues; EXEC ignored) ----
#if defined(KWTA_TDM_LOAD)
  if (tid < 32) {
    kw_u32x4 g0 = kwta_desc_g0(keys_lds_off, (uint64_t)(uintptr_t)rowp);
    kw_i32x8 g1 = kwta_desc_g1();
    kw_i32x4 z4 = {0, 0, 0, 0};
#if defined(KWTA_TDM_6ARG)
    kw_i32x8 z8 = {0, 0, 0, 0, 0, 0, 0, 0};
    __builtin_amdgcn_tensor_load_to_lds(g0, g1, z4, z4, z8, 0);
#else
    __builtin_amdgcn_tensor_load_to_lds(g0, g1, z4, z4, 0);
#endif
    kwta_wait_tensorcnt0();
  }
#else
  for (int i = tid; i < KWTA_E / 4; i += KWTA_T)
    ((float4*)keys)[i] = ((const float4*)rowp)[i];
#endif
  __syncthreads();

  // ---- Convert to monotonic sortable keys (strided, LDS bank-conflict free) ----
  for (int i = tid; i < KWTA_E; i += KWTA_T) {
    uint32_t b = keys[i];
    keys[i] = b ^ ((uint32_t)((int32_t)b >> 31) | 0x80000000u);
  }
  if (tid == 0) { sh_remain = KWTA_K; sh_prefix = 0u; }
  __syncthreads();

  // ---- 4-pass MSB-first radix select for the k-th largest key ----
  uint32_t prefix = 0u;
  for (int shift = 24; shift >= 0; shift -= 8) {
    hist[tid & 255] = 0u;
    __syncthreads();
    const uint32_t pmask = (shift == 24) ? 0u : (0xFFFFFFFFu << (shift + 8));
    for (int i = tid; i < KWTA_E; i += KWTA_T) {
      uint32_t k = keys[i];
      if ((k & pmask) == prefix)
        atomicAdd(&hist[(k >> shift) & 0xFFu], 1u);
    }
    __syncthreads();
    if (tid == 0) {
      uint32_t rem = sh_remain;
      uint32_t b = 255u;
      for (;;) {                 // scan buckets high -> low
        uint32_t c = hist[b];
        if (c >= rem || b == 0u) break;
        rem -= c;
        --b;
      }
      sh_remain = rem;           // rank of target within bucket b
      sh_prefix = sh_prefix | (b << shift);
    }
    __syncthreads();
    prefix = sh_prefix;
  }

  const uint32_t T     = prefix;     // exact key of the k-th largest element
  const uint32_t quota = sh_remain;  // # of ties (==T) to include, lowest index first

  // ---- Exact tie ranking (lower index first, matching lax.top_k) ----
  {
    uint32_t myt = 0;
    const int base_i = tid * KWTA_PT;
    for (int i = 0; i < KWTA_PT; ++i)
      if (keys[base_i + i] == T) ++myt;
    tcnt[tid] = myt;
  }
  __syncthreads();
  if (tid == 0) {                    // serial exclusive scan over 256 entries
    uint32_t run = 0;
    for (int t = 0; t < KWTA_T; ++t) { uint32_t v = tcnt[t]; tcnt[t] = run; run += v; }
  }
  __syncthreads();

  // ---- Write mask bits into LDS in place (chunked, index order) ----
  {
    uint32_t rank = tcnt[tid];
    const int base_i = tid * KWTA_PT;
    for (int i = 0; i < KWTA_PT; ++i) {
      uint32_t k = keys[base_i + i];
      uint32_t v;
      if (k > T) {
        v = 0x3F800000u;                              // 1.0f
      } else if (k == T) {
        v = (rank < quota) ? 0x3F800000u : 0u;
        ++rank;
      } else {
        v = 0u;
      }
      keys[base_i + i] = v;
    }
  }
  __syncthreads();

  // ---- Stream mask to HBM via TDM store (wave 0), else coalesced VGPR path ----
  float* orow = out + (size_t)row * KWTA_E;
#if defined(KWTA_TDM_STORE)
  if (tid < 32) {
    kw_u32x4 g0 = kwta_desc_g0(keys_lds_off, (uint64_t)(uintptr_t)orow);
    kw_i32x8 g1 = kwta_desc_g1();
    kw_i32x4 z4 = {0, 0, 0, 0};
#if defined(KWTA_TDM_6ARG)
    kw_i32x8 z8 = {0, 0, 0, 0, 0, 0, 0, 0};
    __builtin_amdgcn_tensor_store_from_lds(g0, g1, z4, z4, z8, 0);
#else
    __builtin_amdgcn_tensor_store_from_lds(g0, g1, z4, z4, 0);
#endif
    kwta_wait_tensorcnt0();
  }
#else
  for (int i = tid; i < KWTA_E; i += KWTA_T)
    orow[i] = __uint_as_float(keys[i]);
#endif
}

extern "C" void kernel_launch(void* const* d_in, const int* in_sizes, int n_in,
                              void* d_out, int out_size, void* d_ws, size_t ws_size,
                              hipStream_t stream) {
  (void)n_in; (void)out_size; (void)d_ws; (void)ws_size;
  const float* x = (const float*)d_in[0];
  float* out = (float*)d_out;
  const int nrows = in_sizes[0] / KWTA_E;   // 4096
  kwta_kernel<<<dim3(nrows), dim3(KWTA_T), 0, stream>>>(x, out, nrows);
}